// EdgeConvolution_33827162423731
// MI455X (gfx1250) — compile-verified
//
#include <hip/hip_runtime.h>
#include <hip/hip_bf16.h>
#include <float.h>

// ---------------------------------------------------------------------------
// EdgeConv (DGCNN) for MI455X / gfx1250, wave32, WMMA f32 16x16x4.
// Dims fixed by reference: B=8, C=64, N=4096, O=64, K=20.
// ---------------------------------------------------------------------------

typedef float v2f __attribute__((ext_vector_type(2)));
typedef float v8f __attribute__((ext_vector_type(8)));

#define BATCH 8
#define CH    64
#define NPTS  4096
#define OCH   64
#define KNN   20
#define CHUNK 1024
#define NEG_INF (-3.0e38f)

__device__ __forceinline__ v8f wmma_f32(v2f a, v2f b, v8f c) {
    // D = A(16x4, f32) x B(4x16, f32) + C(16x16, f32)
    return __builtin_amdgcn_wmma_f32_16x16x4_f32(
        false, a, false, b, (short)0, c, false, false);
}

// ---------------------------------------------------------------------------
// K1: squared norms per point. sq[b*N + n] = sum_c x[b,c,n]^2
// ---------------------------------------------------------------------------
__global__ __launch_bounds__(256) void sq_kernel(const float* __restrict__ x,
                                                 float* __restrict__ sq) {
    int t = blockIdx.x * 256 + threadIdx.x;   // 0 .. B*N-1
    int b = t >> 12;
    int n = t & (NPTS - 1);
    const float* p = x + (size_t)b * CH * NPTS + n;
    float s = 0.f;
#pragma unroll
    for (int c = 0; c < CH; ++c) { float v = p[(size_t)c * NPTS]; s += v * v; }
    sq[t] = s;
}

// ---------------------------------------------------------------------------
// K2: kNN via WMMA fp32 Gram + register top-20 selection.
// Block: 256 thr (8 waves) handles one (batch, 16-query tile).
// Score s = 2*inner - sq[col]  (same ranking as neg_dist; row term constant).
// ---------------------------------------------------------------------------
__global__ __launch_bounds__(256) void knn_kernel(const float* __restrict__ x,
                                                  const float* __restrict__ sq,
                                                  int* __restrict__ idxout) {
    __shared__ float smem[16 * CHUNK];   // 64 KB score chunk; reused for merge

    const int b    = blockIdx.y;
    const int q0   = blockIdx.x * 16;
    const int tid  = threadIdx.x;
    const int wave = tid >> 5;
    const int lane = tid & 31;
    const int lmod = lane & 15;
    const int khalf  = (lane >> 4) << 1;  // K offset 0 or 2  (A and B identical)
    const int rowadd = (lane >> 4) << 3;  // C rows 0..7 or 8..15
    const float* xb  = x + (size_t)b * CH * NPTS;
    const float* sqb = sq + (size_t)b * NPTS;

    // Preload A (16 queries x 64 channels) into registers: a[kk] covers K=4kk..4kk+3
    v2f a[16];
#pragma unroll
    for (int kk = 0; kk < 16; ++kk) {
        const float* p = xb + (size_t)(4 * kk + khalf) * NPTS + q0 + lmod;
        a[kk].x = p[0];
        a[kk].y = p[NPTS];
    }

    // Per-thread running top-20 (registers; fully unrolled insert -> cndmask)
    float best[KNN]; int bidx[KNN]; float thr = NEG_INF;
#pragma unroll
    for (int i = 0; i < KNN; ++i) { best[i] = NEG_INF; bidx[i] = 0; }

    const int q     = tid >> 4;   // query this thread selects for (0..15)
    const int strip = tid & 15;   // interleaved column strip (bank-friendly)

    for (int c0 = 0; c0 < NPTS; c0 += CHUNK) {
        // ---- GEMM phase: wave w computes col tiles w, w+8, ... of the chunk
        for (int t = wave; t < CHUNK / 16; t += 8) {
            const int col0 = c0 + t * 16;
            v8f acc = {};
#pragma unroll
            for (int kk = 0; kk < 16; ++kk) {
                const float* p = xb + (size_t)(4 * kk + khalf) * NPTS + col0 + lmod;
                v2f bb; bb.x = p[0]; bb.y = p[NPTS];
                acc = wmma_f32(a[kk], bb, acc);
            }
            const float sqc = sqb[col0 + lmod];
#pragma unroll
            for (int r = 0; r < 8; ++r)
                smem[(r + rowadd) * CHUNK + t * 16 + lmod] = 2.0f * acc[r] - sqc;
        }
        __syncthreads();

        // ---- Selection phase: thread scans cols {c0 + i*16 + strip} of row q
        const float* row = smem + q * CHUNK;
        for (int i = 0; i < CHUNK / 16; ++i) {
            float v = row[i * 16 + strip];
            if (v > thr) {
                int col = c0 + i * 16 + strip;
                int mpos = 0; float mval = best[0];
#pragma unroll
                for (int s = 1; s < KNN; ++s)
                    if (best[s] < mval) { mval = best[s]; mpos = s; }
#pragma unroll
                for (int s = 0; s < KNN; ++s)
                    if (s == mpos) { best[s] = v; bidx[s] = col; }
                float nmin = best[0];
#pragma unroll
                for (int s = 1; s < KNN; ++s) nmin = fminf(nmin, best[s]);
                thr = nmin;
            }
        }
        __syncthreads();
    }

    // ---- Merge 16 strips per query (reuse smem: 5120 dists + 5120 idx < 16384)
    int* smi = (int*)smem;
#pragma unroll
    for (int i = 0; i < KNN; ++i) {
        smem[tid * KNN + i]            = best[i];
        smi[16 * CHUNK / 2 + tid * KNN + i] = bidx[i];   // ints at float ofs 8192
    }
    __syncthreads();
    if (tid < 16) {
        float fb[KNN]; int fi[KNN]; float fthr = NEG_INF;
#pragma unroll
        for (int i = 0; i < KNN; ++i) { fb[i] = NEG_INF; fi[i] = 0; }
        for (int s = 0; s < 16; ++s) {
            const int cb = (tid * 16 + s) * KNN;
            for (int i = 0; i < KNN; ++i) {
                float v = smem[cb + i];
                if (v > fthr) {
                    int cidx = smi[16 * CHUNK / 2 + cb + i];
                    int mpos = 0; float mval = fb[0];
#pragma unroll
                    for (int u = 1; u < KNN; ++u)
                        if (fb[u] < mval) { mval = fb[u]; mpos = u; }
#pragma unroll
                    for (int u = 0; u < KNN; ++u)
                        if (u == mpos) { fb[u] = v; fi[u] = cidx; }
                    float nmin = fb[0];
#pragma unroll
                    for (int u = 1; u < KNN; ++u) nmin = fminf(nmin, fb[u]);
                    fthr = nmin;
                }
            }
        }
        int* op = idxout + ((size_t)b * NPTS + q0 + tid) * KNN;
#pragma unroll
        for (int i = 0; i < KNN; ++i) op[i] = fi[i];
    }
}

// ---------------------------------------------------------------------------
// K3: projection GEMMs via WMMA.  y1T[b][n][o] = sum_c W[o][c] x[b][c][n]
//                                 y3T[b][n][o] = sum_c (W[o][64+c]-W[o][c]) x
// One wave per (batch, o-tile, n-tile, which). 16384 wave-tasks.
// ---------------------------------------------------------------------------
__global__ __launch_bounds__(256) void proj_kernel(const float* __restrict__ x,
                                                   const float* __restrict__ W,
                                                   float* __restrict__ y1T,
                                                   float* __restrict__ y3T) {
    int task = blockIdx.x * 8 + (threadIdx.x >> 5);
    int lane = threadIdx.x & 31;
    int lmod = lane & 15;
    int khalf  = (lane >> 4) << 1;
    int rowadd = (lane >> 4) << 3;

    int which = task & 1;
    int nt    = (task >> 1) & 255;
    int ot    = (task >> 9) & 3;
    int b     = task >> 11;

    const float* xb = x + (size_t)b * CH * NPTS;
    const float* wr = W + (size_t)(ot * 16 + lmod) * (2 * CH);
    const int n0 = nt * 16;

    v8f acc = {};
#pragma unroll
    for (int kk = 0; kk < 16; ++kk) {
        int c = 4 * kk + khalf;
        v2f av, bv;
        if (which == 0) { av.x = wr[c];            av.y = wr[c + 1]; }
        else            { av.x = wr[CH + c] - wr[c];
                          av.y = wr[CH + c + 1] - wr[c + 1]; }
        const float* p = xb + (size_t)c * NPTS + n0 + lmod;
        bv.x = p[0]; bv.y = p[NPTS];
        acc = wmma_f32(av, bv, acc);
    }

    float* y = (which ? y3T : y1T)
             + (size_t)b * NPTS * OCH + (size_t)(n0 + lmod) * OCH
             + ot * 16 + rowadd;                 // 32B-aligned
    float4 lo = { acc[0], acc[1], acc[2], acc[3] };
    float4 hi = { acc[4], acc[5], acc[6], acc[7] };
    ((float4*)y)[0] = lo;
    ((float4*)y)[1] = hi;
}

// ---------------------------------------------------------------------------
// K4: gather neighbors, per-point k-reduction (max or min per gamma sign),
//     and per-wave partial BN stats (deterministic, no atomics).
// One wave processes 32 points; lane l owns channels {2l, 2l+1}.
// ---------------------------------------------------------------------------
__global__ __launch_bounds__(256) void gather_kernel(const float* __restrict__ y1T,
                                                     const float* __restrict__ y3T,
                                                     const int* __restrict__ idx,
                                                     const float* __restrict__ gamma,
                                                     float* __restrict__ selbuf,
                                                     float* __restrict__ partial) {
    int waveId = blockIdx.x * 8 + (threadIdx.x >> 5);
    int lane   = threadIdx.x & 31;
    int ch     = lane * 2;
    bool mx0 = gamma[ch] >= 0.f, mx1 = gamma[ch + 1] >= 0.f;

    float s0 = 0.f, s1 = 0.f, q0 = 0.f, q1 = 0.f;

    for (int pt = 0; pt < 32; ++pt) {
        int p = waveId * 32 + pt;
        int b = p >> 12;
        float2 e = ((const float2*)(y3T + (size_t)p * OCH))[lane];
        const int* ip = idx + (size_t)p * KNN;
        const float* y1b = y1T + (size_t)b * NPTS * OCH;

        float hi0 = -FLT_MAX, hi1 = -FLT_MAX, lo0 = FLT_MAX, lo1 = FLT_MAX;
#pragma unroll
        for (int j = 0; j < KNN; ++j) {
            int nb = ip[j];
            float2 g = ((const float2*)(y1b + ((size_t)nb << 6)))[lane];
            float h0 = g.x + e.x, h1 = g.y + e.y;
            s0 += h0; s1 += h1; q0 += h0 * h0; q1 += h1 * h1;
            hi0 = fmaxf(hi0, h0); lo0 = fminf(lo0, h0);
            hi1 = fmaxf(hi1, h1); lo1 = fminf(lo1, h1);
        }
        float2 selv = { mx0 ? hi0 : lo0, mx1 ? hi1 : lo1 };
        ((float2*)(selbuf + (size_t)p * OCH))[lane] = selv;
    }
    float* pb = partial + (size_t)waveId * 128;
    pb[ch]          = s0; pb[ch + 1]       = s1;
    pb[OCH + ch]    = q0; pb[OCH + ch + 1] = q1;
}

// ---------------------------------------------------------------------------
// K5: fixed-order reduction of 1024 wave partials -> per-channel scale/shift
// ---------------------------------------------------------------------------
__global__ void stats_kernel(const float* __restrict__ partial,
                             const float* __restrict__ gamma,
                             const float* __restrict__ beta,
                             float* __restrict__ coeff) {
    int ch = threadIdx.x;
    if (ch >= OCH) return;
    float s = 0.f, q = 0.f;
#pragma unroll 4
    for (int w = 0; w < 1024; ++w) {
        s += partial[(size_t)w * 128 + ch];
        q += partial[(size_t)w * 128 + OCH + ch];
    }
    const float M = (float)BATCH * NPTS * KNN;   // 655360
    float mean  = s / M;
    float var   = q / M - mean * mean;
    float scale = gamma[ch] * rsqrtf(var + 1e-5f);
    coeff[ch]       = scale;
    coeff[OCH + ch] = beta[ch] - mean * scale;
}

// ---------------------------------------------------------------------------
// K6: finalize: out[b][o][n] = LeakyReLU(sel[b][n][o]*scale[o] + shift[o])
// ---------------------------------------------------------------------------
__global__ __launch_bounds__(256) void final_kernel(const float* __restrict__ selbuf,
                                                    const float* __restrict__ coeff,
                                                    float* __restrict__ out) {
    int t = blockIdx.x * 256 + threadIdx.x;      // B*O*N flat, n fastest
    int n = t & (NPTS - 1);
    int o = (t >> 12) & (OCH - 1);
    int b = t >> 18;
    float v = selbuf[(((size_t)b * NPTS + n) << 6) + o];
    float r = v * coeff[o] + coeff[OCH + o];
    out[t] = (r >= 0.f) ? r : 0.2f * r;
}

// ---------------------------------------------------------------------------
extern "C" void kernel_launch(void* const* d_in, const int* in_sizes, int n_in,
                              void* d_out, int out_size, void* d_ws, size_t ws_size,
                              hipStream_t stream) {
    const float* x     = (const float*)d_in[0];   // [8,64,4096]
    const float* W     = (const float*)d_in[1];   // [64,128]
    const float* gamma = (const float*)d_in[2];   // [64]
    const float* beta  = (const float*)d_in[3];   // [64]
    float* out = (float*)d_out;                   // [8,64,4096]

    float* ws    = (float*)d_ws;
    float* sq    = ws;                                  //   32768 f
    int*   idx   = (int*)(ws + 32768);                  //  655360 i
    float* y1T   = ws + 32768 + 655360;                 // 2097152 f
    float* y3T   = y1T + (size_t)BATCH * NPTS * OCH;    // 2097152 f
    float* sel   = y3T + (size_t)BATCH * NPTS * OCH;    // 2097152 f
    float* part  = sel + (size_t)BATCH * NPTS * OCH;    //  131072 f
    float* coeff = part + 1024 * 128;                   //     128 f
    // total ~28.5 MB

    sq_kernel<<<(BATCH * NPTS) / 256, 256, 0, stream>>>(x, sq);
    knn_kernel<<<dim3(NPTS / 16, BATCH), 256, 0, stream>>>(x, sq, idx);
    proj_kernel<<<2048, 256, 0, stream>>>(x, W, y1T, y3T);
    gather_kernel<<<128, 256, 0, stream>>>(y1T, y3T, idx, gamma, sel, part);
    stats_kernel<<<1, 64, 0, stream>>>(part, gamma, beta, coeff);
    final_kernel<<<(BATCH * OCH * NPTS) / 256, 256, 0, stream>>>(sel, coeff, out);
}